// NeuralLorenz96_2156073582655
// MI455X (gfx1250) — compile-verified
//
#include <hip/hip_runtime.h>

// Two-scale Lorenz-96, RK4, sequential scan of NT-1 steps.
// Single persistent workgroup; state in registers + LDS candidate buffer.
// History rows streamed out by the CDNA5 Tensor Data Mover
// (tensor_store_from_lds, TENSORcnt); coupling rows prefetched with
// async LDS loads (global_load_async_to_lds_b32, ASYNCcnt).

#define NT   2048
#define KK   256
#define JJ   32
#define KJ   8192   // KK*JJ
#define NTHR 1024

typedef unsigned int v4u __attribute__((ext_vector_type(4)));
typedef int          v4i __attribute__((ext_vector_type(4)));
typedef int          v8i __attribute__((ext_vector_type(8)));

// Issue one TDM store: a 1-row 2D tile of `nelem` f32 from LDS to global.
// D# layout per CDNA5 ISA ch.8 (group0: count/lds_addr/global_addr/type,
// group1: data_size/dims/strides; groups 2-3 zero for a <=2D tensor).
__device__ __forceinline__ void tdm_store_row(unsigned lds_base,
                                              const float* gptr,
                                              unsigned nelem)
{
    unsigned long long ga = (unsigned long long)gptr;
    v4u g0;
    g0[0] = 1u;                                   // count=1, is_restore=0, gather off
    g0[1] = lds_base;                             // lds_addr (bytes)
    g0[2] = (unsigned)ga;                         // global_addr[31:0]
    g0[3] = (unsigned)(ga >> 32) | (2u << 30);    // global_addr[56:32] | type=2

    v8i g1;
    g1[0] = 0x20000;                              // data_size=2 (4 bytes); mask=0
    g1[1] = (int)((nelem & 0xFFFFu) << 16);       // tensor_dim0[15:0]
    g1[2] = (int)(((nelem >> 16) & 0xFFFFu) | (1u << 16)); // td0[31:16] | tensor_dim1=1
    g1[3] = (int)((nelem & 0xFFFFu) << 16);       // td1[31:16]=0 | tile_dim0=nelem
    g1[4] = 1;                                    // tile_dim1=1 | tile_dim2=0
    g1[5] = (int)nelem;                           // tensor_dim0_stride[31:0]
    g1[6] = 0;                                    // stride0[47:32] | stride1[15:0]
    g1[7] = 0;                                    // stride1[47:16]

    v4i gz = {0, 0, 0, 0};
#if defined(__clang_major__) && (__clang_major__ >= 23)
    v8i gz8 = {0, 0, 0, 0, 0, 0, 0, 0};
    __builtin_amdgcn_tensor_store_from_lds(g0, g1, gz, gz, gz8, 0);
#else
    __builtin_amdgcn_tensor_store_from_lds(g0, g1, gz, gz, 0);
#endif
}

__global__ __launch_bounds__(NTHR, 1)
void lorenz96_rk4_kernel(const float* __restrict__ X0,
                         const float* __restrict__ Y0,
                         const float* __restrict__ coupling,
                         const float* __restrict__ pF,
                         const float* __restrict__ ph,
                         const float* __restrict__ pc,
                         const float* __restrict__ pdt,
                         float* __restrict__ out)
{
    __shared__ float Cx[KK];          // candidate X (also final state staging)
    __shared__ float Cy[KJ];          // candidate Y (also final state staging)
    __shared__ float CoupBuf[2][KK];  // double-buffered coupling rows

    const int t = (int)threadIdx.x;

    const float F  = pF[0];
    const float h  = ph[0];
    const float c  = pc[0];
    const float dt = pdt[0];
    const float cJ     = c * (float)JJ;
    const float hcJ    = h * c * (1.0f / (float)JJ);
    const float halfdt = 0.5f * dt;
    const float dt6    = dt * (1.0f / 6.0f);

    float* const outX = out;                    // [NT, KK]
    float* const outY = out + (size_t)NT * KK;  // [NT, KJ]

    // 32-bit LDS byte offsets (low 32 bits of generic pointer)
    const unsigned ldsCyBase = (unsigned)(unsigned long long)(void*)&Cy[0];
    const unsigned ldsCxBase = (unsigned)(unsigned long long)(void*)&Cx[0];
    const unsigned ldsCp0 = (unsigned)(unsigned long long)(void*)&CoupBuf[0][(t < KK) ? t : 0];
    const unsigned ldsCp1 = (unsigned)(unsigned long long)(void*)&CoupBuf[1][(t < KK) ? t : 0];

    // ---- load initial state (row 0 of X0 / Y0) ----
    float sx = 0.0f;
    if (t < KK) { sx = X0[t]; Cx[t] = sx; }
    float sy[8];
#pragma unroll
    for (int i = 0; i < 8; ++i) { sy[i] = Y0[8 * t + i]; Cy[8 * t + i] = sy[i]; }

    // Async-prefetch coupling row 0 into buffer 0 (used by step n=1).
    if (t < KK) {
        const float* g = coupling + t;
        asm volatile("global_load_async_to_lds_b32 %0, %1, off"
                     :: "v"(ldsCp0), "v"(g) : "memory");
    }

    __syncthreads();   // all initial LDS writes visible before TDM reads

    // History row 0 via TDM (wave 0 only; TDM ignores EXEC, so one wave issues).
    if (t < 32) {
        tdm_store_row(ldsCyBase, outY, KJ);
        tdm_store_row(ldsCxBase, outX, KK);
    }

    float dx = 0.0f, dy[8], w[11];

    // Derivative at the candidate state held in Cx/Cy.
    auto deriv = [&](float coup) {
        if (t < KK) {
            const float xm1 = Cx[(t - 1) & (KK - 1)];
            const float xp1 = Cx[(t + 1) & (KK - 1)];
            const float xm2 = Cx[(t - 2) & (KK - 1)];
            dx = xm1 * (xp1 - xm2) - Cx[t] + F + coup;
        }
        const int base = 8 * t;
#pragma unroll
        for (int i = 0; i < 11; ++i) w[i] = Cy[(base - 1 + i) & (KJ - 1)];
        const float cxv = Cx[t >> 2];   // X[(8t+i)/32] constant over i=0..7
#pragma unroll
        for (int i = 0; i < 8; ++i)
            dy[i] = -cJ * w[i + 2] * (w[i + 3] - w[i]) - c * w[i + 1] + hcJ * cxv;
    };

    auto writeC = [&](float a) {
        if (t < KK) Cx[t] = fmaf(a, dx, sx);
#pragma unroll
        for (int i = 0; i < 8; ++i) Cy[8 * t + i] = fmaf(a, dy[i], sy[i]);
    };

    for (int n = 1; n < NT; ++n) {
        // Wave 0: prior-step TDM reads of C done (we overwrite C this step).
        __builtin_amdgcn_s_wait_tensorcnt((short)0);
        // Waves 0-7: this step's coupling row has landed in LDS.
        asm volatile("s_wait_asynccnt 0" ::: "memory");
        __syncthreads();

        const int buf = (n - 1) & 1;
        const float coup = (t < KK) ? CoupBuf[buf][t] : 0.0f;

        // Kick off async load of next step's coupling row into the other buffer.
        if ((t < KK) && (n < NT - 1)) {
            const float* g = coupling + (size_t)n * KK + t;
            const unsigned dst = (n & 1) ? ldsCp1 : ldsCp0;
            asm volatile("global_load_async_to_lds_b32 %0, %1, off"
                         :: "v"(dst), "v"(g) : "memory");
        }

        float accx, accy[8];

        // ---- stage 1: d1 at S (C currently holds S) ----
        deriv(coup);
        accx = dx;
#pragma unroll
        for (int i = 0; i < 8; ++i) accy[i] = dy[i];
        __syncthreads();
        writeC(halfdt);                 // C = S + dt/2 * d1
        __syncthreads();

        // ---- stage 2 ----
        deriv(coup);
        accx += 2.0f * dx;
#pragma unroll
        for (int i = 0; i < 8; ++i) accy[i] += 2.0f * dy[i];
        __syncthreads();
        writeC(halfdt);                 // C = S + dt/2 * d2
        __syncthreads();

        // ---- stage 3 ----
        deriv(coup);
        accx += 2.0f * dx;
#pragma unroll
        for (int i = 0; i < 8; ++i) accy[i] += 2.0f * dy[i];
        __syncthreads();
        writeC(dt);                     // C = S + dt * d3
        __syncthreads();

        // ---- stage 4 ----
        deriv(coup);
        accx += dx;
#pragma unroll
        for (int i = 0; i < 8; ++i) accy[i] += dy[i];
        __syncthreads();                // all stage-4 reads of C done

        // ---- RK4 combine; publish new state into C (state for next step) ----
        sx = fmaf(dt6, accx, sx);
#pragma unroll
        for (int i = 0; i < 8; ++i) sy[i] = fmaf(dt6, accy[i], sy[i]);
        if (t < KK) Cx[t] = sx;
#pragma unroll
        for (int i = 0; i < 8; ++i) Cy[8 * t + i] = sy[i];
        __syncthreads();                // state visible in LDS for TDM + next stage-1

        // Stream history row n out of LDS via the Tensor Data Mover; next
        // step's stage-1 only READS C, so compute overlaps the DMA.
        if (t < 32) {
            tdm_store_row(ldsCyBase, outY + (size_t)n * KJ, KJ);
            tdm_store_row(ldsCxBase, outX + (size_t)n * KK, KK);
        }
    }

    // Drain outstanding DMA / async ops (S_ENDPGM also implies wait-idle).
    __builtin_amdgcn_s_wait_tensorcnt((short)0);
    asm volatile("s_wait_asynccnt 0" ::: "memory");
}

extern "C" void kernel_launch(void* const* d_in, const int* in_sizes, int n_in,
                              void* d_out, int out_size, void* d_ws, size_t ws_size,
                              hipStream_t stream) {
    (void)in_sizes; (void)n_in; (void)out_size; (void)d_ws; (void)ws_size;
    const float* X0   = (const float*)d_in[0];
    const float* Y0   = (const float*)d_in[1];
    const float* coup = (const float*)d_in[2];
    const float* F    = (const float*)d_in[3];
    const float* h    = (const float*)d_in[4];
    // d_in[5] = b, unused by the math (as in the reference)
    const float* c    = (const float*)d_in[6];
    const float* dt   = (const float*)d_in[7];

    lorenz96_rk4_kernel<<<dim3(1), dim3(NTHR), 0, stream>>>(
        X0, Y0, coup, F, h, c, dt, (float*)d_out);
}